// Dependency_GAN_53970559042185
// MI455X (gfx1250) — compile-verified
//
#include <hip/hip_runtime.h>

// Problem constants
#define B_      2
#define L_      384
#define H_      768
#define HHALF_  384
#define EMB_    64
#define NE_     384
#define NTYPES_ 76

typedef float v2f __attribute__((ext_vector_type(2)));
typedef float v4f __attribute__((ext_vector_type(4)));
typedef float v8f __attribute__((ext_vector_type(8)));

// ---------------------------------------------------------------------------
// Graph scatter: dense [B,L,L] int graph, duplicates summed (matches .at[].add)
// ---------------------------------------------------------------------------
__global__ void zero_graph_kernel(int* __restrict__ g, int n) {
    for (int i = blockIdx.x * blockDim.x + threadIdx.x; i < n;
         i += gridDim.x * blockDim.x)
        g[i] = 0;
}

__global__ void scatter_kernel(const int* __restrict__ coo,
                               const int* __restrict__ et,
                               int* __restrict__ g) {
    int idx = blockIdx.x * blockDim.x + threadIdx.x;
    if (idx >= B_ * NE_) return;
    int b = idx / NE_, e = idx % NE_;
    int r = coo[b * 2 * NE_ + e];            // coo[b,0,e]
    int c = coo[b * 2 * NE_ + NE_ + e];      // coo[b,1,e]
    atomicAdd(&g[b * L_ * L_ + r * L_ + c], et[b * NE_ + e]);
}

// dep_graphs (as float) + dep_masks outputs
__global__ void graph_finalize_kernel(const int* __restrict__ g,
                                      float* __restrict__ out_graph,
                                      float* __restrict__ out_mask, int n) {
    for (int i = blockIdx.x * blockDim.x + threadIdx.x; i < n;
         i += gridDim.x * blockDim.x) {
        int v = g[i];
        out_graph[i] = (float)v;
        out_mask[i]  = (v > 0) ? 0.0f : -10000.0f;
    }
}

// Per-edge-type score: tscore[t] = dot(etype_emb[t], w_att[384:448])
__global__ void tscore_kernel(const float* __restrict__ emb,
                              const float* __restrict__ w_att,
                              float* __restrict__ ts) {
    int t = threadIdx.x;
    if (t < NTYPES_) {
        float s = 0.0f;
        for (int e = 0; e < EMB_; ++e)
            s += emb[t * EMB_ + e] * w_att[HHALF_ + e];
        ts[t] = s;
    }
}

// ---------------------------------------------------------------------------
// Q/K projection GEMM: [B*L,768] @ [768,384] (+bias), one 16x16 tile per wave
// fp32 WMMA 16x16x4. 2304 tiles total (Q tiles then K tiles).
// ---------------------------------------------------------------------------
__global__ void qk_wmma_kernel(const float* __restrict__ hidden,
                               const float* __restrict__ Wq, const float* __restrict__ bq,
                               const float* __restrict__ Wk, const float* __restrict__ bk,
                               float* __restrict__ Qb, float* __restrict__ Kb) {
    const int MT = (B_ * L_) / 16;   // 48
    const int NT = HHALF_ / 16;      // 24
    int wave = blockIdx.x * (blockDim.x >> 5) + (threadIdx.x >> 5);
    int lane = threadIdx.x & 31;
    int which = wave / (MT * NT);        // 0 = Q, 1 = K
    int t     = wave % (MT * NT);
    int mt = t / NT, nt = t % NT;
    const float* W    = which ? Wk : Wq;
    const float* bias = which ? bk : bq;
    float*       Out  = which ? Kb : Qb;
    int row0 = mt * 16, col0 = nt * 16;
    int m  = lane & 15;            // A: row M; B: col N; C: col N
    int kb = (lane >> 4) << 1;     // K sub-offset: 0 or 2
    v8f c = {};
    for (int k0 = 0; k0 < H_; k0 += 4) {
        v2f a, b;
        a.x = hidden[(row0 + m) * H_ + k0 + kb];
        a.y = hidden[(row0 + m) * H_ + k0 + kb + 1];
        // B[k,n] = W[col0+n, k]   (W is [384,768], Q = hidden @ W^T)
        b.x = W[(col0 + m) * H_ + k0 + kb];
        b.y = W[(col0 + m) * H_ + k0 + kb + 1];
        c = __builtin_amdgcn_wmma_f32_16x16x4_f32(false, a, false, b,
                                                  (short)0, c, false, false);
    }
    float bv = bias[col0 + m];
    int hi = (lane >> 4) * 8;
    for (int v = 0; v < 8; ++v)
        Out[(row0 + v + hi) * HHALF_ + col0 + m] = c[v] + bv;
}

// ---------------------------------------------------------------------------
// elmwise_hidden[b,i,j,:] = q_i * k_j   -- 453 MB pure write stream.
// One block per (b*L+i, group of 8 j). q row in LDS, k rows from L2,
// 16B non-temporal stores (never re-read; don't pollute L2).
// ---------------------------------------------------------------------------
__global__ void elmwise_kernel(const float* __restrict__ Qb,
                               const float* __restrict__ Kb,
                               float* __restrict__ out) {
    __shared__ float qrow[HHALF_] __attribute__((aligned(16)));
    int bi  = blockIdx.y;                    // b*L + i
    int jg  = blockIdx.x;                    // j group: 8 columns
    int tid = threadIdx.x;
    for (int d = tid; d < HHALF_; d += blockDim.x)
        qrow[d] = Qb[bi * HHALF_ + d];
    __syncthreads();
    int b = bi / L_;
    const float* Kp = Kb + (size_t)b * L_ * HHALF_;
    float* op = out + (size_t)bi * L_ * HHALF_;
    const int NV4 = HHALF_ / 4;              // 96 float4 per (i,j)
    for (int item = tid; item < 8 * NV4; item += blockDim.x) {
        int jj = item / NV4;
        int d4 = item % NV4;
        int j  = jg * 8 + jj;
        v4f kd = ((const v4f*)(Kp + (size_t)j * HHALF_))[d4];
        v4f qd = *(const v4f*)(&qrow[d4 * 4]);
        v4f r  = qd * kd;
        __builtin_nontemporal_store(r, (v4f*)(op + (size_t)j * HHALF_) + d4);
    }
}

// ---------------------------------------------------------------------------
// Scores: S[b,i,j] = (Q_i . (w1 * K_j)) + tscore[clip(g)] + b_att + mask
// i.e. (Q * w1) @ K^T as fp32 WMMA GEMM, fused epilogue. 1152 tiles.
// ---------------------------------------------------------------------------
__global__ void score_wmma_kernel(const float* __restrict__ Qb,
                                  const float* __restrict__ Kb,
                                  const float* __restrict__ w_att,
                                  const float* __restrict__ b_att,
                                  const int* __restrict__ g,
                                  const float* __restrict__ ts,
                                  float* __restrict__ scores) {
    const int NT = L_ / 16;                   // 24
    const int TPB = NT * NT;                  // 576 tiles per batch
    int wave = blockIdx.x * (blockDim.x >> 5) + (threadIdx.x >> 5);
    int lane = threadIdx.x & 31;
    int b = wave / TPB;
    int t = wave % TPB;
    int it = t / NT, jt = t % NT;
    const float* Qp = Qb + (size_t)b * L_ * HHALF_;
    const float* Kp = Kb + (size_t)b * L_ * HHALF_;
    int i0 = it * 16, j0 = jt * 16;
    int m  = lane & 15;
    int kb = (lane >> 4) << 1;
    v8f c = {};
    for (int k0 = 0; k0 < HHALF_; k0 += 4) {
        v2f a, bb;
        a.x = Qp[(i0 + m) * HHALF_ + k0 + kb]     * w_att[k0 + kb];
        a.y = Qp[(i0 + m) * HHALF_ + k0 + kb + 1] * w_att[k0 + kb + 1];
        bb.x = Kp[(j0 + m) * HHALF_ + k0 + kb];       // B[k,j] = K[j,k]
        bb.y = Kp[(j0 + m) * HHALF_ + k0 + kb + 1];
        c = __builtin_amdgcn_wmma_f32_16x16x4_f32(false, a, false, bb,
                                                  (short)0, c, false, false);
    }
    float ba = b_att[0];
    int hi = (lane >> 4) * 8;
    for (int v = 0; v < 8; ++v) {
        int i = i0 + v + hi, j = j0 + m;
        int gv = g[b * L_ * L_ + i * L_ + j];
        int gi = gv < 0 ? 0 : (gv > NTYPES_ - 1 ? NTYPES_ - 1 : gv);  // jnp.take clips
        float mask = (gv > 0) ? 0.0f : -10000.0f;
        scores[((size_t)b * L_ + i) * L_ + j] = c[v] + ts[gi] + ba + mask;
    }
}

// ---------------------------------------------------------------------------
// Row softmax over 384 columns, in place. One 128-thread block per (b,i).
// ---------------------------------------------------------------------------
__global__ void softmax_kernel(float* __restrict__ scores) {
    __shared__ float red[128];
    int row = blockIdx.x, tid = threadIdx.x;
    float* p = scores + (size_t)row * L_;
    float lmax = -3.4e38f;
    for (int j = tid; j < L_; j += 128) lmax = fmaxf(lmax, p[j]);
    red[tid] = lmax; __syncthreads();
    for (int s = 64; s > 0; s >>= 1) {
        if (tid < s) red[tid] = fmaxf(red[tid], red[tid + s]);
        __syncthreads();
    }
    float mx = red[0]; __syncthreads();
    float lsum = 0.0f;
    for (int j = tid; j < L_; j += 128) {
        float e = __expf(p[j] - mx);
        p[j] = e;
        lsum += e;
    }
    red[tid] = lsum; __syncthreads();
    for (int s = 64; s > 0; s >>= 1) {
        if (tid < s) red[tid] += red[tid + s];
        __syncthreads();
    }
    float inv = 1.0f / red[0];
    for (int j = tid; j < L_; j += 128) p[j] *= inv;
}

// ---------------------------------------------------------------------------
// output = P @ hidden : [384,384]@[384,768] per batch, fp32 WMMA. 2304 tiles.
// ---------------------------------------------------------------------------
__global__ void out_wmma_kernel(const float* __restrict__ P,
                                const float* __restrict__ hidden,
                                float* __restrict__ out) {
    const int MT = L_ / 16;                   // 24
    const int NT = H_ / 16;                   // 48
    int wave = blockIdx.x * (blockDim.x >> 5) + (threadIdx.x >> 5);
    int lane = threadIdx.x & 31;
    int b = wave / (MT * NT);
    int t = wave % (MT * NT);
    int mt = t / NT, nt = t % NT;
    const float* Pp = P + (size_t)b * L_ * L_;
    const float* Hp = hidden + (size_t)b * L_ * H_;
    int i0 = mt * 16, n0 = nt * 16;
    int m  = lane & 15;
    int kb = (lane >> 4) << 1;
    v8f c = {};
    for (int k0 = 0; k0 < L_; k0 += 4) {
        v2f a, bb;
        a.x = Pp[(i0 + m) * L_ + k0 + kb];
        a.y = Pp[(i0 + m) * L_ + k0 + kb + 1];
        bb.x = Hp[(k0 + kb) * H_ + n0 + m];       // B row-major
        bb.y = Hp[(k0 + kb + 1) * H_ + n0 + m];
        c = __builtin_amdgcn_wmma_f32_16x16x4_f32(false, a, false, bb,
                                                  (short)0, c, false, false);
    }
    int hi = (lane >> 4) * 8;
    for (int v = 0; v < 8; ++v)
        out[((size_t)b * L_ + i0 + v + hi) * H_ + n0 + m] = c[v];
}

// ---------------------------------------------------------------------------
extern "C" void kernel_launch(void* const* d_in, const int* in_sizes, int n_in,
                              void* d_out, int out_size, void* d_ws, size_t ws_size,
                              hipStream_t stream) {
    (void)in_sizes; (void)n_in; (void)out_size; (void)ws_size;
    const float* hidden = (const float*)d_in[0];
    const int*   coo    = (const int*)d_in[1];
    const int*   etype  = (const int*)d_in[2];
    const float* Wq     = (const float*)d_in[3];
    const float* bq     = (const float*)d_in[4];
    const float* Wk     = (const float*)d_in[5];
    const float* bk     = (const float*)d_in[6];
    const float* watt   = (const float*)d_in[7];
    const float* batt   = (const float*)d_in[8];
    const float* eemb   = (const float*)d_in[9];

    // Output tuple, concatenated flat: output, elmwise_hidden, dep_graphs, dep_masks
    float* out        = (float*)d_out;
    float* out_output = out;
    float* out_elm    = out + (size_t)B_ * L_ * H_;
    float* out_graph  = out_elm + (size_t)B_ * L_ * L_ * HHALF_;
    float* out_mask   = out_graph + (size_t)B_ * L_ * L_;

    // Workspace: Q, K, scores (fp32), type-score table, int graph
    float* Qb = (float*)d_ws;
    float* Kb = Qb + (size_t)B_ * L_ * HHALF_;
    float* Sc = Kb + (size_t)B_ * L_ * HHALF_;
    float* Ts = Sc + (size_t)B_ * L_ * L_;
    int*   G  = (int*)(Ts + 128);

    zero_graph_kernel<<<288, 256, 0, stream>>>(G, B_ * L_ * L_);
    scatter_kernel<<<(B_ * NE_ + 255) / 256, 256, 0, stream>>>(coo, etype, G);
    tscore_kernel<<<1, 128, 0, stream>>>(eemb, watt, Ts);
    qk_wmma_kernel<<<288, 256, 0, stream>>>(hidden, Wq, bq, Wk, bk, Qb, Kb);
    graph_finalize_kernel<<<288, 256, 0, stream>>>(G, out_graph, out_mask, B_ * L_ * L_);
    elmwise_kernel<<<dim3(L_ / 8, B_ * L_), 256, 0, stream>>>(Qb, Kb, out_elm);
    score_wmma_kernel<<<144, 256, 0, stream>>>(Qb, Kb, watt, batt, G, Ts, Sc);
    softmax_kernel<<<B_ * L_, 128, 0, stream>>>(Sc);
    out_wmma_kernel<<<288, 256, 0, stream>>>(Sc, hidden, out_output);
}